// GATv2Layer_12979391169461
// MI455X (gfx1250) — compile-verified
//
#include <hip/hip_runtime.h>

#define Nn 2048
#define Ff 128
#define HEADS 8
#define OUT_DIM 8
#define HID (HEADS * OUT_DIM) /* 64 */
#define ALPHA 0.2f
#define NEG_BIG -1000000000.0f

typedef __attribute__((ext_vector_type(2))) float v2f;
typedef __attribute__((ext_vector_type(4))) float v4f;
typedef __attribute__((ext_vector_type(8))) float v8f;

__device__ __forceinline__ float lrelu(float x) { return x > 0.f ? x : ALPHA * x; }

// ---------------------------------------------------------------------------
// Kernel 1: h = X (2048x128) @ W (128x64) via V_WMMA_F32_16X16X4_F32.
// One wave per 16x16 output tile; 4 waves/block cover all 64 columns.
// A-frag (16x4 f32): lanes 0-15 -> M=lane, K={0,1}; lanes 16-31 -> M=lane-16, K={2,3}
// B-frag (4x16 f32): mirrored; C/D: VGPR v -> M = v (+8 for lanes>=16), N = lane&15.
// ---------------------------------------------------------------------------
__global__ __launch_bounds__(128) void gemm_h_kernel(const float* __restrict__ X,
                                                     const float* __restrict__ W,
                                                     float* __restrict__ h) {
    const int wave = threadIdx.x >> 5;
    const int lane = threadIdx.x & 31;
    const int rowBase = blockIdx.x * 16;
    const int colBase = wave * 16;
    const int r = lane & 15;
    const int kq = (lane >> 4) * 2;

    v8f c = {};
    const float* xrow = X + (rowBase + r) * Ff;
#pragma unroll 4
    for (int kk = 0; kk < Ff; kk += 4) {
        v2f a = *(const v2f*)(xrow + kk + kq);       // K = kk+kq, kk+kq+1 (8B aligned)
        v2f b;
        b.x = W[(kk + kq) * HID + colBase + r];
        b.y = W[(kk + kq + 1) * HID + colBase + r];
        c = __builtin_amdgcn_wmma_f32_16x16x4_f32(
            /*neg_a=*/false, a, /*neg_b=*/false, b,
            /*c_mod=*/(short)0, c, /*reuse_a=*/false, /*reuse_b=*/false);
    }
    const int mAdd = (lane >> 4) * 8;
#pragma unroll
    for (int v = 0; v < 8; ++v)
        h[(rowBase + v + mAdd) * HID + colBase + r] = c[v];
}

// ---------------------------------------------------------------------------
// Kernel 2: s1[n,j] = sum_d h[n,j,d]*a1[d] ; s2[n,j] = sum_d h[n,j,d]*a2[d]
// ---------------------------------------------------------------------------
__global__ void scores_kernel(const float* __restrict__ h, const float* __restrict__ ak,
                              float* __restrict__ s1, float* __restrict__ s2) {
    int n = blockIdx.x * blockDim.x + threadIdx.x;
    if (n >= Nn) return;
    const float* hn = h + n * HID;
#pragma unroll
    for (int j = 0; j < HEADS; ++j) {
        float acc1 = 0.f, acc2 = 0.f;
#pragma unroll
        for (int d = 0; d < OUT_DIM; ++d) {
            float v = hn[j * OUT_DIM + d];
            acc1 += v * ak[d];
            acc2 += v * ak[OUT_DIM + d];
        }
        s1[n * HEADS + j] = acc1;
        s2[n * HEADS + j] = acc2;
    }
}

// ---------------------------------------------------------------------------
// Kernel 3: one block per row a. Online-softmax over b (axis 1), then the
// 8x8 matvec against h[a] and a fully-coalesced 32B/thread output stream.
// Output (134 MB, write-once) uses non-temporal stores so it doesn't evict
// the reused A rows / s2 table from the 192 MB L2.
// ---------------------------------------------------------------------------
__global__ __launch_bounds__(256) void gat_attend_kernel(const int* __restrict__ A,
                                                         const float* __restrict__ h,
                                                         const float* __restrict__ s1,
                                                         const float* __restrict__ s2,
                                                         float* __restrict__ out) {
    const int n = blockIdx.x;
    const int t = threadIdx.x;
    const int lane = t & 31;
    const int wave = t >> 5;

    __shared__ float hs[HID];
    __shared__ float redM[8][HEADS];
    __shared__ float redL[8][HEADS];
    __shared__ float Mf[HEADS];
    __shared__ float Rd[HEADS];

    if (t < HID) hs[t] = h[n * HID + t];

    float s1r[HEADS];
#pragma unroll
    for (int j = 0; j < HEADS; ++j) s1r[j] = s1[n * HEADS + j];

    const int* Arow = A + (size_t)n * Nn;

    float m[HEADS], l[HEADS];
#pragma unroll
    for (int j = 0; j < HEADS; ++j) { m[j] = -__builtin_inff(); l[j] = 0.f; }

    // Pass 1: per-thread online softmax stats over strided b.
    for (int i = t; i < Nn; i += 256) {
        const int a = Arow[i];
        v4f s2lo = *(const v4f*)(s2 + i * HEADS);
        v4f s2hi = *(const v4f*)(s2 + i * HEADS + 4);
#pragma unroll
        for (int j = 0; j < HEADS; ++j) {
            float sv = (j < 4) ? s2lo[j] : s2hi[j - 4];
            float e = (a > 0) ? lrelu(s1r[j] + sv) : NEG_BIG;
            float M = fmaxf(m[j], e);
            l[j] = l[j] * __expf(m[j] - M) + __expf(e - M);
            m[j] = M;
        }
    }

    // Wave32 reduction.
#pragma unroll
    for (int j = 0; j < HEADS; ++j) {
        for (int off = 16; off > 0; off >>= 1) {
            float mo = __shfl_xor(m[j], off, 32);
            float lo = __shfl_xor(l[j], off, 32);
            float M = fmaxf(m[j], mo);
            l[j] = l[j] * __expf(m[j] - M) + lo * __expf(mo - M);
            m[j] = M;
        }
        if (lane == 0) { redM[wave][j] = m[j]; redL[wave][j] = l[j]; }
    }
    __syncthreads();

    // Cross-wave reduction: thread j (<8) owns head j.
    if (t < HEADS) {
        float M = redM[0][t], L = redL[0][t];
#pragma unroll
        for (int w = 1; w < 8; ++w) {
            float mo = redM[w][t], lo = redL[w][t];
            float Mx = fmaxf(M, mo);
            L = L * __expf(M - Mx) + lo * __expf(mo - Mx);
            M = Mx;
        }
        Mf[t] = M;
        Rd[t] = 1.0f / L;
    }
    __syncthreads();

    float Mr[HEADS], Rr[HEADS];
#pragma unroll
    for (int j = 0; j < HEADS; ++j) { Mr[j] = Mf[j]; Rr[j] = Rd[j]; }

    // Pass 2: attention weights + 8x8 matvec + streamed NT b128 stores.
    for (int i = t; i < Nn; i += 256) {
        const int a = Arow[i];
        v4f s2lo = *(const v4f*)(s2 + i * HEADS);
        v4f s2hi = *(const v4f*)(s2 + i * HEADS + 4);
        float p[HEADS];
#pragma unroll
        for (int j = 0; j < HEADS; ++j) {
            float sv = (j < 4) ? s2lo[j] : s2hi[j - 4];
            float e = (a > 0) ? lrelu(s1r[j] + sv) : NEG_BIG;
            p[j] = __expf(e - Mr[j]) * Rr[j];
        }
        v4f o0, o1;
#pragma unroll
        for (int d = 0; d < OUT_DIM; ++d) {
            float acc = 0.f;
#pragma unroll
            for (int j = 0; j < HEADS; ++j) acc += p[j] * hs[j * OUT_DIM + d];
            acc = lrelu(acc);
            if (d < 4) o0[d] = acc; else o1[d - 4] = acc;
        }
        float* dst = out + ((size_t)n * Nn + i) * OUT_DIM;
        __builtin_nontemporal_store(o0, (v4f*)(dst));
        __builtin_nontemporal_store(o1, (v4f*)(dst + 4));
    }
}

extern "C" void kernel_launch(void* const* d_in, const int* in_sizes, int n_in,
                              void* d_out, int out_size, void* d_ws, size_t ws_size,
                              hipStream_t stream) {
    const float* X = (const float*)d_in[0];        // (2048,128)
    const int* A = (const int*)d_in[1];            // (2048,2048)
    const float* W = (const float*)d_in[2];        // (128,64)
    const float* ak = (const float*)d_in[3];       // (16,1)
    float* out = (float*)d_out;                    // (2048*2048*8,)

    float* h = (float*)d_ws;                       // 2048*64
    float* s1 = h + Nn * HID;                      // 2048*8
    float* s2 = s1 + Nn * HEADS;                   // 2048*8

    gemm_h_kernel<<<Nn / 16, 128, 0, stream>>>(X, W, h);
    scores_kernel<<<(Nn + 255) / 256, 256, 0, stream>>>(h, ak, s1, s2);
    gat_attend_kernel<<<Nn, 256, 0, stream>>>(A, h, s1, s2, out);
}